// MappedTransposedConvolution_58463094833213
// MI455X (gfx1250) — compile-verified
//
#include <hip/hip_runtime.h>

typedef __attribute__((ext_vector_type(2))) float v2f;
typedef __attribute__((ext_vector_type(8))) float v8f;

// Problem constants (match reference)
constexpr int B_   = 2;
constexpr int CIN  = 32;
constexpr int COUT = 32;
constexpr int KK   = 9;
constexpr int IH   = 256, IW = 256;
constexpr int OH   = 512, OW = 512;
constexpr int P    = IH * IW;      // 65536 input pixels
constexpr int OP   = OH * OW;      // 262144 output pixels (2^18)
constexpr int NT   = 16;           // pixels per wave tile (WMMA N)
constexpr int PT   = P / NT;       // 4096 pixel tiles
constexpr int WPB  = 8;            // waves per 256-thread block

__device__ __forceinline__ void atomAddF32(float* p, float v) {
  unsafeAtomicAdd(p, v);           // native global_atomic_add_f32 (no return)
}

// Fill output with broadcast bias so the splat kernel can accumulate in place.
__global__ __launch_bounds__(256)
void bias_init_kernel(const float* __restrict__ bias, float* __restrict__ out) {
  size_t i = (size_t)blockIdx.x * 256 + threadIdx.x;   // total B*COUT*OP
  out[i] = bias[(i >> 18) & (COUT - 1)];               // (i / OP) % COUT
}

// One wave handles (batch bb, tap k, 16-pixel tile):
//   GEMM  : contrib[o, n] = sum_i w[i,o,k] * x[bb,i,pix0+n]  via v_wmma_f32_16x16x4_f32
//   SPLAT : bilinear scatter of contrib into out[bb, o, :] with fp32 atomics
__global__ __launch_bounds__(256)
void mtc_kernel(const float* __restrict__ x,      // [B, CIN, P]
                const float* __restrict__ w,      // [CIN, COUT, KK]
                const float* __restrict__ sm,     // [P, KK, 2]
                float* __restrict__ out) {        // [B, COUT, OP]
  const int lane = threadIdx.x & 31;
  const int wave = threadIdx.x >> 5;
  const int unit = blockIdx.x * WPB + wave;       // 0 .. PT*KK*B_-1 (all waves full)

  const int pt   = unit & (PT - 1);
  const int rest = unit >> 12;                    // / PT
  const int k    = rest % KK;
  const int bb   = rest / KK;
  const int pix0 = pt * NT;

  const int n   = lane & 15;                      // WMMA column (pixel within tile)
  const int hi  = lane >> 4;                      // lane half selects K-pair / M-half
  const int pix = pix0 + n;

  // ---------------- GEMM via WMMA f32 16x16x4, CIN=32 -> 8 K-steps ----------
  // A (16x4, M=cout K=cin): lanes 0-15 hold K rows {0,1}, lanes 16-31 rows {2,3}
  // B (4x16, K=cin N=pix) : same K split, lane&15 = N
  // D (16x16)             : lane&15 = N, VGPR r = M (lanes 16-31: M = 8+r)
  v8f d0 = {};                                    // couts 0..15
  v8f d1 = {};                                    // couts 16..31
  const float* xb = x + (size_t)bb * CIN * P;
  const int kp = hi * 2;
#pragma unroll
  for (int s = 0; s < CIN / 4; ++s) {
    const int i0 = s * 4 + kp;                    // this lane-half's first K row
    v2f bf;
    bf.x = xb[(size_t)i0 * P + pix];
    bf.y = xb[(size_t)(i0 + 1) * P + pix];
    const float* wr0 = w + (size_t)i0 * (COUT * KK) + n * KK + k;   // A[m=n]
    v2f a0, a1;
    a0.x = wr0[0];
    a0.y = wr0[COUT * KK];                        // K row i0+1
    const float* wr1 = wr0 + 16 * KK;             // A[m=16+n]
    a1.x = wr1[0];
    a1.y = wr1[COUT * KK];
    d0 = __builtin_amdgcn_wmma_f32_16x16x4_f32(false, a0, false, bf,
                                               (short)0, d0, false, false);
    d1 = __builtin_amdgcn_wmma_f32_16x16x4_f32(false, a1, false, bf,
                                               (short)0, d1, false, false);
  }

  // ---------------- bilinear splat coefficients for (k, pix) ----------------
  const float* smp = sm + (size_t)pix * (KK * 2) + k * 2;
  const float sx = smp[0];
  const float sy = smp[1];
  const float x0f = floorf(sx), y0f = floorf(sy);
  const float dx = sx - x0f, dy = sy - y0f;
  int x0 = (int)x0f; x0 = x0 < 0 ? 0 : (x0 > OW - 1 ? OW - 1 : x0);
  int y0 = (int)y0f; y0 = y0 < 0 ? 0 : (y0 > OH - 1 ? OH - 1 : y0);
  const int x1 = x0 + 1 > OW - 1 ? OW - 1 : x0 + 1;
  const int y1 = y0 + 1 > OH - 1 ? OH - 1 : y0 + 1;
  const float ws[4] = { (1.f - dx) * (1.f - dy), dx * (1.f - dy),
                        (1.f - dx) * dy,         dx * dy };
  const int idx[4]  = { y0 * OW + x0, y0 * OW + x1,
                        y1 * OW + x0, y1 * OW + x1 };

  // ---------------- scatter: each lane holds pixel n, couts mlo+r (+16) -----
  float* outb = out + (size_t)bb * COUT * OP;
  const int mlo = hi * 8;
#pragma unroll
  for (int c = 0; c < 4; ++c) {
    const float wc = ws[c];
    float* po = outb + idx[c];
#pragma unroll
    for (int r = 0; r < 8; ++r) {
      atomAddF32(po + (size_t)(mlo + r) * OP,      d0[r] * wc);
      atomAddF32(po + (size_t)(16 + mlo + r) * OP, d1[r] * wc);
    }
  }
}

extern "C" void kernel_launch(void* const* d_in, const int* in_sizes, int n_in,
                              void* d_out, int out_size, void* d_ws, size_t ws_size,
                              hipStream_t stream) {
  const float* x    = (const float*)d_in[0];   // [B, CIN, IH, IW]
  const float* wgt  = (const float*)d_in[1];   // [CIN, COUT, KK]
  const float* bias = (const float*)d_in[2];   // [COUT]
  const float* sm   = (const float*)d_in[3];   // [IH, IW, KK, 2]
  float* out        = (float*)d_out;           // [B, COUT, OH, OW]

  const size_t total = (size_t)B_ * COUT * OP; // 16,777,216
  bias_init_kernel<<<(int)(total / 256), 256, 0, stream>>>(bias, out);

  const int units  = PT * KK * B_;             // 73728 wave-units
  const int blocks = units / WPB;              // 9216 blocks, 8 full waves each
  mtc_kernel<<<blocks, 256, 0, stream>>>(x, wgt, sm, out);
}